// BoxHead_61675730370639
// MI455X (gfx1250) — compile-verified
//
#include <hip/hip_runtime.h>
#include <stdint.h>

// ---------------------------------------------------------------------------
// Faster R-CNN box-head postprocess for MI455X (gfx1250, wave32)
// ---------------------------------------------------------------------------

namespace {
constexpr int   kB     = 8;
constexpr int   kN     = 4000;
constexpr int   kC     = 91;
constexpr int   kCm1   = 90;
constexpr int   kPre   = 2048;   // K_PRE
constexpr int   kCap   = 16384;  // per-image candidate cap (power of 2, bitonic)
constexpr int   kDet   = 100;
constexpr float kThresh = 0.05f;
constexpr float kNms    = 0.5f;
constexpr float kXClip  = 4.135166556742356f; // ln(1000/16)
}

// ---- gfx1250 async-to-LDS path (guarded; falls back to plain loads) -------
#if defined(__gfx1250__) && \
    __has_builtin(__builtin_amdgcn_global_load_async_to_lds_b128) && \
    __has_builtin(__builtin_amdgcn_s_wait_asynccnt)
#define HAVE_ASYNC_LDS 1
typedef int v4i_t __attribute__((vector_size(16)));
__device__ __forceinline__ __attribute__((address_space(1))) v4i_t*
to_glb(const void* p) {
  return (__attribute__((address_space(1))) v4i_t*)p;
}
__device__ __forceinline__ __attribute__((address_space(3))) v4i_t*
to_lds(void* p) {
  return (__attribute__((address_space(3))) v4i_t*)p;
}
#else
#define HAVE_ASYNC_LDS 0
#endif

// ---------------------------------------------------------------------------
// torchvision BoxCoder.decode + clip (single proposal, single class)
// ---------------------------------------------------------------------------
__device__ __forceinline__ void decode_clip(const float pr[4], const float rg[4],
                                            float fw, float fh, float out[4]) {
  float w  = pr[2] - pr[0];
  float h  = pr[3] - pr[1];
  float cx = pr[0] + 0.5f * w;
  float cy = pr[1] + 0.5f * h;
  float dx = rg[0] / 10.0f;
  float dy = rg[1] / 10.0f;
  float dw = fminf(rg[2] / 5.0f, kXClip);
  float dh = fminf(rg[3] / 5.0f, kXClip);
  float pcx = dx * w + cx;
  float pcy = dy * h + cy;
  float pw  = expf(dw) * w;
  float ph  = expf(dh) * h;
  out[0] = fminf(fmaxf(pcx - 0.5f * pw, 0.0f), fw);
  out[1] = fminf(fmaxf(pcy - 0.5f * ph, 0.0f), fh);
  out[2] = fminf(fmaxf(pcx + 0.5f * pw, 0.0f), fw);
  out[3] = fminf(fmaxf(pcy + 0.5f * ph, 0.0f), fh);
}

// ---------------------------------------------------------------------------
// K0: zero per-image candidate counters (graph-replay safe re-init)
// ---------------------------------------------------------------------------
__global__ void k_zero(int* cand_cnt) {
  if (threadIdx.x < kB) cand_cnt[threadIdx.x] = 0;
}

// ---------------------------------------------------------------------------
// K1: streaming softmax + decode-filter.  One wave32 per proposal row.
// ---------------------------------------------------------------------------
__global__ void __launch_bounds__(256)
k_score(const float* __restrict__ logits, const float* __restrict__ reg,
        const float* __restrict__ props, const int* __restrict__ ihp,
        const int* __restrict__ iwp, uint32_t* __restrict__ cand_key,
        uint32_t* __restrict__ cand_idx, int* __restrict__ cand_cnt) {
  int wave = (int)((blockIdx.x * blockDim.x + threadIdx.x) >> 5);
  int lane = (int)(threadIdx.x & 31);
  if (wave >= kB * kN) return;
  int img = wave / kN;
  int n   = wave - img * kN;

  const float* lrow = logits + (size_t)wave * kC;
  const float* rrow = reg + (size_t)wave * (4 * kC);
  __builtin_prefetch(rrow, 0, 0);  // global_prefetch_b8 on gfx1250

  float fw = (float)iwp[0];
  float fh = (float)ihp[0];

  // lanes cover classes {lane, lane+32, lane+64}
  float l0 = lrow[lane];
  float l1 = lrow[lane + 32];
  bool  has2 = (lane + 64) < kC;
  float l2 = has2 ? lrow[lane + 64] : -3.4e38f;

  float m = fmaxf(l0, fmaxf(l1, l2));
  for (int s = 16; s > 0; s >>= 1) m = fmaxf(m, __shfl_xor(m, s, 32));

  float e0 = expf(l0 - m);
  float e1 = expf(l1 - m);
  float e2 = has2 ? expf(l2 - m) : 0.0f;
  float sum = e0 + e1 + e2;
  for (int s = 16; s > 0; s >>= 1) sum += __shfl_xor(sum, s, 32);
  float inv = 1.0f / sum;

  float pr[4];
  pr[0] = props[(size_t)wave * 4 + 0];
  pr[1] = props[(size_t)wave * 4 + 1];
  pr[2] = props[(size_t)wave * 4 + 2];
  pr[3] = props[(size_t)wave * 4 + 3];

  float es[3] = {e0, e1, e2};
#pragma unroll
  for (int t = 0; t < 3; ++t) {
    int c = lane + 32 * t;
    if (c < 1 || c >= kC) continue;  // drop background + OOB
    float score = es[t] * inv;
    if (score > kThresh) {
      const float* rp = rrow + 4 * c;
      float rg[4] = {rp[0], rp[1], rp[2], rp[3]};
      float bx[4];
      decode_clip(pr, rg, fw, fh, bx);
      if ((bx[2] - bx[0]) >= 1.0f && (bx[3] - bx[1]) >= 1.0f) {
        int pos = atomicAdd(&cand_cnt[img], 1);
        if (pos < kCap) {
          cand_key[(size_t)img * kCap + pos] = __float_as_uint(score);
          cand_idx[(size_t)img * kCap + pos] = (uint32_t)(n * kCm1 + (c - 1));
        }
      }
    }
  }
}

// ---------------------------------------------------------------------------
// K2: per-image bitonic sort (descending) of packed (score,~idx) keys in
//     128 KB dynamic LDS (CDNA5: 320 KB/WGP), then decode the top 2048.
// ---------------------------------------------------------------------------
__global__ void __launch_bounds__(1024)
k_sort(const uint32_t* __restrict__ cand_key, const uint32_t* __restrict__ cand_idx,
       const int* __restrict__ cand_cnt, const float* __restrict__ reg,
       const float* __restrict__ props, const int* __restrict__ ihp,
       const int* __restrict__ iwp, float* __restrict__ top_score,
       float* __restrict__ top_box, int* __restrict__ top_label,
       float* __restrict__ maxcoord) {
  extern __shared__ unsigned long long skey[];  // kCap entries = 128 KB
  __shared__ float red[32];
  int img = (int)blockIdx.x;
  int tid = (int)threadIdx.x;

  int cnt = cand_cnt[img];
  if (cnt > kCap) cnt = kCap;

  for (int e = tid; e < kCap; e += 1024) {
    unsigned long long v = 0ull;
    if (e < cnt) {
      v = ((unsigned long long)cand_key[(size_t)img * kCap + e] << 32) |
          (uint32_t)(~cand_idx[(size_t)img * kCap + e]);  // ~idx: ties -> smaller idx first
    }
    skey[e] = v;
  }
  __syncthreads();

  // Bitonic sort, descending.
  for (int k = 2; k <= kCap; k <<= 1) {
    for (int j = k >> 1; j >= 1; j >>= 1) {
      for (int i = tid; i < kCap; i += 1024) {
        int l = i ^ j;
        if (l > i) {
          unsigned long long a = skey[i];
          unsigned long long b = skey[l];
          bool up = (i & k) == 0;
          if (up ? (a < b) : (a > b)) { skey[i] = b; skey[l] = a; }
        }
      }
      __syncthreads();
    }
  }

  float fw = (float)iwp[0];
  float fh = (float)ihp[0];
  float locmax = -3.4e38f;
  for (int t = tid; t < kPre; t += 1024) {
    unsigned long long v = skey[t];
    float score = __uint_as_float((uint32_t)(v >> 32));
    uint32_t idx = ~(uint32_t)v;
    float bx[4] = {0.f, 0.f, 0.f, 0.f};
    int lab = 0;
    if (score > kThresh) {
      int n = (int)(idx / kCm1);
      int c = (int)(idx - (uint32_t)n * kCm1) + 1;
      int row = img * kN + n;
      float pr[4] = {props[(size_t)row * 4 + 0], props[(size_t)row * 4 + 1],
                     props[(size_t)row * 4 + 2], props[(size_t)row * 4 + 3]};
      const float* rp = reg + (size_t)row * (4 * kC) + 4 * c;
      float rg[4] = {rp[0], rp[1], rp[2], rp[3]};
      decode_clip(pr, rg, fw, fh, bx);
      lab = c;
      locmax = fmaxf(locmax, fmaxf(fmaxf(bx[0], bx[1]), fmaxf(bx[2], bx[3])));
    }
    top_score[(size_t)img * kPre + t] = (score > kThresh) ? score : 0.0f;
    top_label[(size_t)img * kPre + t] = lab;
    float* tb = top_box + ((size_t)img * kPre + t) * 4;
    tb[0] = bx[0]; tb[1] = bx[1]; tb[2] = bx[2]; tb[3] = bx[3];
  }
  // block max-reduce for per-class offset (jnp.max(tb))
  for (int s = 16; s > 0; s >>= 1) locmax = fmaxf(locmax, __shfl_xor(locmax, s, 32));
  if ((tid & 31) == 0) red[tid >> 5] = locmax;
  __syncthreads();
  if (tid < 32) {
    float v = red[tid];
    for (int s = 16; s > 0; s >>= 1) v = fmaxf(v, __shfl_xor(v, s, 32));
    if (tid == 0) maxcoord[img] = v;
  }
}

// ---------------------------------------------------------------------------
// K3: suppression bitmask matrix [2048][32] u64 per image.  Boxes staged to
//     LDS via gfx1250 async-to-LDS (each box is reused 2048x).
//     grid = (B, 2048/256), block = 256; thread -> one row i.
// ---------------------------------------------------------------------------
__global__ void __launch_bounds__(256)
k_iou(const float* __restrict__ top_box, const int* __restrict__ top_label,
      const float* __restrict__ maxcoord, unsigned long long* __restrict__ supp) {
  __shared__ float sbox[kPre * 4];   // 32 KB
  __shared__ float sarea[kPre];      //  8 KB
  __shared__ int   slab[kPre];       //  8 KB
  int img = (int)blockIdx.x;
  int tid = (int)threadIdx.x;

  const float* gb = top_box + (size_t)img * kPre * 4;
  const int*   gl = top_label + (size_t)img * kPre;

#if HAVE_ASYNC_LDS
#pragma unroll
  for (int k = 0; k < 8; ++k) {  // 2048*4 floats, 16B per request, coalesced
    int e = (k * 256 + tid) * 4;
    __builtin_amdgcn_global_load_async_to_lds_b128(to_glb(gb + e), to_lds(sbox + e), 0, 0);
  }
#pragma unroll
  for (int k = 0; k < 2; ++k) {  // 2048 ints
    int e = (k * 256 + tid) * 4;
    __builtin_amdgcn_global_load_async_to_lds_b128(to_glb(gl + e), to_lds(slab + e), 0, 0);
  }
  __builtin_amdgcn_s_wait_asynccnt(0);
#else
#pragma unroll
  for (int k = 0; k < 8; ++k) {
    int e = (k * 256 + tid) * 4;
    *(float4*)(sbox + e) = *(const float4*)(gb + e);
  }
#pragma unroll
  for (int k = 0; k < 2; ++k) {
    int e = (k * 256 + tid) * 4;
    *(int4*)(slab + e) = *(const int4*)(gl + e);
  }
#endif
  __syncthreads();

  float off_scale = maxcoord[img] + 1.0f;  // batched_nms class offset
  for (int e = tid; e < kPre; e += 256) {
    float off = (float)slab[e] * off_scale;
    float x0 = sbox[e * 4 + 0] + off;
    float y0 = sbox[e * 4 + 1] + off;
    float x1 = sbox[e * 4 + 2] + off;
    float y1 = sbox[e * 4 + 3] + off;
    sbox[e * 4 + 0] = x0; sbox[e * 4 + 1] = y0;
    sbox[e * 4 + 2] = x1; sbox[e * 4 + 3] = y1;
    sarea[e] = (x1 - x0) * (y1 - y0);
  }
  __syncthreads();

  int i = (int)blockIdx.y * 256 + tid;
  float ax0 = sbox[i * 4 + 0], ay0 = sbox[i * 4 + 1];
  float ax1 = sbox[i * 4 + 2], ay1 = sbox[i * 4 + 3];
  float aa  = sarea[i];
  unsigned long long* srow = supp + ((size_t)img * kPre + i) * 32;

  for (int w = 0; w < 32; ++w) {
    unsigned long long bits = 0ull;
    int jbase = w * 64;
    if (jbase + 63 > i) {
      for (int jj = 0; jj < 64; ++jj) {
        int j = jbase + jj;
        if (j <= i) continue;
        float lx = fmaxf(ax0, sbox[j * 4 + 0]);
        float ly = fmaxf(ay0, sbox[j * 4 + 1]);
        float rx = fminf(ax1, sbox[j * 4 + 2]);
        float ry = fminf(ay1, sbox[j * 4 + 3]);
        float iw = fmaxf(rx - lx, 0.0f);
        float ih = fmaxf(ry - ly, 0.0f);
        float inter = iw * ih;
        float iou = inter / (aa + sarea[j] - inter);  // NaN -> no suppression
        if (iou > kNms) bits |= (1ull << jj);
      }
    }
    srow[w] = bits;
  }
}

// ---------------------------------------------------------------------------
// K4: greedy NMS scan + top-100 emit.  One wave32 per image; lane j owns
//     suppression word j, broadcast via __shfl (wave32-correct).
// ---------------------------------------------------------------------------
__global__ void __launch_bounds__(32)
k_nms(const float* __restrict__ top_score, const float* __restrict__ top_box,
      const int* __restrict__ top_label, const unsigned long long* __restrict__ supp,
      float* __restrict__ out_boxes, float* __restrict__ out_scores,
      int* __restrict__ out_labels) {
  int img  = (int)blockIdx.x;
  int lane = (int)threadIdx.x;

  // init all output slots (harness poisons d_out)
  for (int j = lane; j < kDet; j += 32) {
    out_scores[img * kDet + j] = 0.0f;
    out_labels[img * kDet + j] = -1;
    float* ob = out_boxes + (size_t)(img * kDet + j) * 4;
    ob[0] = 0.f; ob[1] = 0.f; ob[2] = 0.f; ob[3] = 0.f;
  }

  unsigned long long rem = 0ull;    // lane's suppressed-word
  unsigned long long keepw = 0ull;  // lane's keep-word
  for (int i = 0; i < kPre; ++i) {
    int wi = i >> 6;
    unsigned long long rw = __shfl(rem, wi, 32);
    bool suppressed = (rw >> (i & 63)) & 1ull;
    float s = top_score[(size_t)img * kPre + i];  // uniform load
    bool keep = (!suppressed) && (s > kThresh);
    if (keep) {
      rem |= supp[((size_t)img * kPre + i) * 32 + lane];
      if (lane == wi) keepw |= (1ull << (i & 63));
    }
  }

  // emit first 100 kept (already in descending-score order)
  int cnt = 0;
  for (int i = 0; i < kPre && cnt < kDet; ++i) {
    unsigned long long kw = __shfl(keepw, i >> 6, 32);
    if ((kw >> (i & 63)) & 1ull) {
      if (lane < 4)
        out_boxes[(size_t)(img * kDet + cnt) * 4 + lane] =
            top_box[((size_t)img * kPre + i) * 4 + lane];
      if (lane == 4) out_scores[img * kDet + cnt] = top_score[(size_t)img * kPre + i];
      if (lane == 5) out_labels[img * kDet + cnt] = top_label[(size_t)img * kPre + i];
      ++cnt;
    }
  }
}

// ---------------------------------------------------------------------------
// host launcher
// ---------------------------------------------------------------------------
extern "C" void kernel_launch(void* const* d_in, const int* in_sizes, int n_in,
                              void* d_out, int out_size, void* d_ws, size_t ws_size,
                              hipStream_t stream) {
  (void)in_sizes; (void)n_in; (void)out_size; (void)ws_size;
  const float* logits = (const float*)d_in[0];
  const float* reg    = (const float*)d_in[1];
  const float* props  = (const float*)d_in[2];
  const int*   ih     = (const int*)d_in[3];
  const int*   iw     = (const int*)d_in[4];

  char* w = (char*)d_ws;
  auto carve = [&](size_t bytes) {
    char* p = w;
    w += (bytes + 255) & ~(size_t)255;
    return (void*)p;
  };
  uint32_t* cand_key = (uint32_t*)carve((size_t)kB * kCap * 4);
  uint32_t* cand_idx = (uint32_t*)carve((size_t)kB * kCap * 4);
  int*      cand_cnt = (int*)carve((size_t)kB * 4);
  float*    top_score = (float*)carve((size_t)kB * kPre * 4);
  float*    top_box   = (float*)carve((size_t)kB * kPre * 16);
  int*      top_label = (int*)carve((size_t)kB * kPre * 4);
  float*    maxcoord  = (float*)carve((size_t)kB * 4);
  unsigned long long* supp =
      (unsigned long long*)carve((size_t)kB * kPre * 32 * 8);

  float* out_boxes  = (float*)d_out;
  float* out_scores = out_boxes + (size_t)kB * kDet * 4;
  int*   out_labels = (int*)(out_scores + (size_t)kB * kDet);

  k_zero<<<1, 32, 0, stream>>>(cand_cnt);
  k_score<<<(kB * kN + 7) / 8, 256, 0, stream>>>(logits, reg, props, ih, iw,
                                                 cand_key, cand_idx, cand_cnt);
  k_sort<<<kB, 1024, kCap * sizeof(unsigned long long), stream>>>(
      cand_key, cand_idx, cand_cnt, reg, props, ih, iw, top_score, top_box,
      top_label, maxcoord);
  k_iou<<<dim3(kB, kPre / 256), 256, 0, stream>>>(top_box, top_label, maxcoord, supp);
  k_nms<<<kB, 32, 0, stream>>>(top_score, top_box, top_label, supp, out_boxes,
                               out_scores, out_labels);
}